// Model_5660766896137
// MI455X (gfx1250) — compile-verified
//
#include <hip/hip_runtime.h>
#include <stdint.h>

// grid_nms for N=8192 points on gfx1250 (MI455X).
//   out[0:8192]     = keep mask in ORIGINAL order (0/1 as float)
//   out[8192:16384] = sorted scores * keep_sorted
//
// Phases: bitonic sort (LDS) -> pairwise suppression bitmask via
// V_WMMA_F32_16X16X4_F32 (dist^2 = ni + nj - 2*dot) -> single-wave
// register-resident sequential scan -> scatter/finalize.

#define NPTS   8192
#define NWORDS 256            // 8192 bits / 32 per row
#define RAD2   64.0f          // radius 8 squared

typedef float v2f __attribute__((ext_vector_type(2)));
typedef float v8f __attribute__((ext_vector_type(8)));

// ---------------------------------------------------------------------------
// Kernel 1: bitonic sort of (score desc, idx asc) in LDS; emit sorted arrays.
// ---------------------------------------------------------------------------
__global__ __launch_bounds__(1024)
void nms_sort(const float* __restrict__ coords, const float* __restrict__ scores,
              float* __restrict__ sscore, int* __restrict__ sidx,
              float* __restrict__ sx, float* __restrict__ sy,
              float* __restrict__ snorm) {
    __shared__ float key[NPTS];   // 32 KB
    __shared__ int   val[NPTS];   // 32 KB
    const unsigned tid = threadIdx.x;

    for (unsigned t = tid; t < NPTS; t += 1024) { key[t] = scores[t]; val[t] = (int)t; }

    for (unsigned k = 2; k <= NPTS; k <<= 1) {
        for (unsigned j = k >> 1; j > 0; j >>= 1) {
            __syncthreads();
            for (unsigned t = tid; t < NPTS / 2; t += 1024) {
                unsigned i = ((t & ~(j - 1u)) << 1) | (t & (j - 1u));
                unsigned x = i | j;
                float ki = key[i], kx = key[x];
                int   vi = val[i], vx = val[x];
                // "x should precede i" in final (desc score, asc idx) order
                bool before_x_i = (kx > ki) || (kx == ki && vx < vi);
                bool dir = (i & k) == 0;
                if (dir == before_x_i) {
                    key[i] = kx; key[x] = ki; val[i] = vx; val[x] = vi;
                }
            }
        }
    }
    __syncthreads();

    for (unsigned r = tid; r < NPTS; r += 1024) {
        int o = val[r];
        float x = coords[2 * o]     - 256.0f;   // center to reduce ni+nj-2dot cancellation
        float y = coords[2 * o + 1] - 256.0f;
        sscore[r] = key[r];
        sidx[r]   = o;
        sx[r]     = x;
        sy[r]     = y;
        snorm[r]  = x * x + y * y;
    }
}

// ---------------------------------------------------------------------------
// Kernel 2: suppression bitmask. One wave computes a 16(row) x 32(col) tile:
// two V_WMMA_F32_16X16X4_F32 (K=2 padded to 4 with zero lanes 16..31),
// then 16 wave32 ballots pack the dist^2 < 64 predicates into mask words.
// mask[i*256 + w] bit b  <=>  dist(i, 32w+b) < radius.
// ---------------------------------------------------------------------------
__global__ __launch_bounds__(256)
void nms_mask(const float* __restrict__ sx, const float* __restrict__ sy,
              const float* __restrict__ snorm, uint32_t* __restrict__ mask) {
    const unsigned lane = threadIdx.x & 31u;
    const unsigned gw   = blockIdx.x * (blockDim.x >> 5) + (threadIdx.x >> 5);
    const unsigned wt   = gw & (NWORDS - 1);   // column word  [0,256)
    const unsigned rt   = gw >> 8;             // row tile     [0,512)
    const unsigned i0   = rt << 4;
    const unsigned j0   = wt << 5;
    if (j0 + 31u <= i0) return;                // pure lower triangle: never read by scan

    const bool     hi  = lane >= 16;           // lanes 16..31 carry K=2,3 (zero-padded)
    const unsigned l15 = lane & 15u;

    // A (16x4 f32): lanes 0-15 = rows M=lane, vgpr0=K0(x), vgpr1=K1(y); hi lanes zero.
    v2f a;
    a.x = hi ? 0.0f : sx[i0 + l15];
    a.y = hi ? 0.0f : sy[i0 + l15];
    // B (4x16 f32): lanes 0-15 = cols N=lane, vgpr0=K0(x), vgpr1=K1(y); hi lanes zero.
    v2f bA, bB;
    bA.x = hi ? 0.0f : sx[j0 + l15];
    bA.y = hi ? 0.0f : sy[j0 + l15];
    bB.x = hi ? 0.0f : sx[j0 + 16 + l15];
    bB.y = hi ? 0.0f : sy[j0 + 16 + l15];

    v8f z = {0.f, 0.f, 0.f, 0.f, 0.f, 0.f, 0.f, 0.f};
    // D = A x B : D[i][j] = xi*xj + yi*yj
    v8f cA = __builtin_amdgcn_wmma_f32_16x16x4_f32(false, a, false, bA, (short)0, z, false, false);
    v8f cB = __builtin_amdgcn_wmma_f32_16x16x4_f32(false, a, false, bB, (short)0, z, false, false);

    // Row norms for this lane's C rows: vgpr k holds row (k + (hi?8:0)).
    float ni[8];
#pragma unroll
    for (int k = 0; k < 8; ++k) ni[k] = snorm[i0 + (hi ? 8u : 0u) + k];
    const float njA = snorm[j0 + l15];
    const float njB = snorm[j0 + 16 + l15];

    // Ballot per C vgpr: bits[15:0] = row k cols 0..15, bits[31:16] = row k+8 cols 0..15.
    uint32_t ba[8], bb[8];
#pragma unroll
    for (int k = 0; k < 8; ++k) {
        ba[k] = __builtin_amdgcn_ballot_w32((ni[k] + njA - 2.0f * cA[k]) < RAD2);
        bb[k] = __builtin_amdgcn_ballot_w32((ni[k] + njB - 2.0f * cB[k]) < RAD2);
    }

    if (lane == 0) {  // ballot results are wave-uniform; 16 scalar stores
#pragma unroll
        for (int r = 0; r < 8; ++r) {
            mask[(size_t)(i0 + r)     * NWORDS + wt] = (ba[r] & 0xFFFFu) | ((bb[r] & 0xFFFFu) << 16);
            mask[(size_t)(i0 + r + 8) * NWORDS + wt] = (ba[r] >> 16)     | (bb[r] & 0xFFFF0000u);
        }
    }
}

// ---------------------------------------------------------------------------
// Kernel 3: sequential greedy scan, single wave32, keep bits in registers
// (8 words/lane x 32 lanes = 8192 bits). No barriers; readlane broadcast.
// ---------------------------------------------------------------------------
__global__ __launch_bounds__(32)
void nms_scan(const uint32_t* __restrict__ mask, uint32_t* __restrict__ keepw) {
    const unsigned lane = threadIdx.x;   // 0..31, lane owns words [lane*8, lane*8+8)
    uint32_t kw[8];
#pragma unroll
    for (int q = 0; q < 8; ++q) kw[q] = 0xFFFFFFFFu;

    for (int i = 0; i < NPTS; ++i) {
        const unsigned wi    = (unsigned)i >> 5;
        const unsigned owner = wi >> 3;
        const unsigned qi    = wi & 7u;
        const unsigned bit   = (unsigned)i & 31u;

        uint32_t v = kw[0];
#pragma unroll
        for (int q = 1; q < 8; ++q) v = (qi == (unsigned)q) ? kw[q] : v;
        uint32_t rk = (uint32_t)__builtin_amdgcn_readlane((int)v, (int)owner);

        if ((rk >> bit) & 1u) {          // point i survives: suppress all j>i within radius
            const uint32_t* row = mask + (size_t)i * NWORDS + lane * 8u;
#pragma unroll
            for (int q = 0; q < 8; ++q) {
                const unsigned wg = lane * 8u + (unsigned)q;
                uint32_t m = row[q];
                const uint32_t low = (2u << bit) - 1u;        // bits [0..bit] (bit=31 -> all)
                m = (wg == wi) ? (m & ~low) : m;              // only j > i in boundary word
                m = (wg <  wi) ? 0u : m;                      // j < word(i): untouched
                kw[q] &= ~m;
            }
        }
    }
#pragma unroll
    for (int q = 0; q < 8; ++q) keepw[lane * 8u + q] = kw[q];
}

// ---------------------------------------------------------------------------
// Kernel 4: finalize. keep_orig (original order, via sidx permutation scatter)
// and suppressed_scores (sorted order).
// ---------------------------------------------------------------------------
__global__ __launch_bounds__(256)
void nms_final(const float* __restrict__ sscore, const int* __restrict__ sidx,
               const uint32_t* __restrict__ keepw, float* __restrict__ out) {
    const unsigned r = blockIdx.x * 256u + threadIdx.x;
    if (r >= NPTS) return;
    const uint32_t bit = (keepw[r >> 5] >> (r & 31u)) & 1u;
    out[sidx[r]]    = (float)bit;                  // keep mask, original order
    out[NPTS + r]   = bit ? sscore[r] : 0.0f;      // surviving scores, sorted order
}

// ---------------------------------------------------------------------------
extern "C" void kernel_launch(void* const* d_in, const int* in_sizes, int n_in,
                              void* d_out, int out_size, void* d_ws, size_t ws_size,
                              hipStream_t stream) {
    const float* coords = (const float*)d_in[0];   // [8192,2] f32
    const float* scores = (const float*)d_in[1];   // [8192]   f32
    float*       out    = (float*)d_out;           // [16384]  f32

    char* ws = (char*)d_ws;
    float*    sscore = (float*)(ws + 0);            // 32 KB
    float*    sx     = (float*)(ws + 32768);        // 32 KB
    float*    sy     = (float*)(ws + 65536);        // 32 KB
    float*    snorm  = (float*)(ws + 98304);        // 32 KB
    int*      sidx   = (int*)  (ws + 131072);       // 32 KB
    uint32_t* keepw  = (uint32_t*)(ws + 163840);    // 1 KB
    uint32_t* mask   = (uint32_t*)(ws + (1 << 18)); // 8 MB @ 256 KB offset

    nms_sort <<<1, 1024, 0, stream>>>(coords, scores, sscore, sidx, sx, sy, snorm);

    // 512 row-tiles x 256 col-words = 131072 wave-tiles, 8 waves per 256-thread block
    nms_mask <<<16384, 256, 0, stream>>>(sx, sy, snorm, mask);

    nms_scan <<<1, 32, 0, stream>>>(mask, keepw);

    nms_final<<<32, 256, 0, stream>>>(sscore, sidx, keepw, out);
}